// CapsuleLayer_12025908428989
// MI455X (gfx1250) — compile-verified
//
#include <hip/hip_runtime.h>

typedef float v2f __attribute__((ext_vector_type(2)));
typedef float v8f __attribute__((ext_vector_type(8)));

#define IN_NODES  1152
#define OUT_NODES 10
#define IN_DIM    8
#define OUT_DIM   16
#define BATCH     256
#define SVOL      (OUT_NODES * BATCH * OUT_DIM)   // 40960 elements per s buffer

// ---------------- init b logits to zero ----------------
__global__ void k_init(float* __restrict__ b) {
    int i = blockIdx.x * blockDim.x + threadIdx.x;
    if (i < IN_NODES * OUT_NODES) b[i] = 0.0f;
}

// ---------------- softmax over out-nodes per in-node ----------------
__global__ void k_softmax(const float* __restrict__ b, float* __restrict__ c) {
    int i = blockIdx.x * blockDim.x + threadIdx.x;
    if (i >= IN_NODES) return;
    float row[OUT_NODES];
    float m = -1e30f;
    #pragma unroll
    for (int j = 0; j < OUT_NODES; ++j) { row[j] = b[i * OUT_NODES + j]; m = fmaxf(m, row[j]); }
    float sum = 0.0f;
    #pragma unroll
    for (int j = 0; j < OUT_NODES; ++j) { row[j] = __expf(row[j] - m); sum += row[j]; }
    float inv = 1.0f / sum;
    #pragma unroll
    for (int j = 0; j < OUT_NODES; ++j) c[i * OUT_NODES + j] = row[j] * inv;
}

// ---------------- s[j,b,d] = sum_i c[i,j] * (x[b,i,:] @ W[i,j,d,:]) ----------------
// Grid: 160 blocks = j(10) x mt2(8 pairs of batch tiles) x ihalf(2).
// Block = 256 thr = 8 waves; each wave reduces 1152/16 = 72 in-nodes and owns TWO
// 16x16 accumulators (batch rows m0..m0+15 and m0+16..m0+31) so the scaled-W B-operand
// is fetched once per 4 wmma.  Partials over the i-halves land in two s buffers,
// summed deterministically by k_squash.
__global__ __launch_bounds__(256) void k_s(const float* __restrict__ x,
                                           const float* __restrict__ W,
                                           const float* __restrict__ c,
                                           float* __restrict__ s_part) {
    __shared__ float red[8 * 2 * 256];                 // 16 KB
    const int j     = blockIdx.x / 16;
    const int rsub  = blockIdx.x % 16;
    const int mt2   = rsub >> 1;                       // 0..7 -> 32 batch rows
    const int ihalf = rsub & 1;
    const int m0    = mt2 * 32;
    const int wave  = threadIdx.x >> 5;
    const int lane  = threadIdx.x & 31;
    const int l     = lane & 15;
    const int h     = lane >> 4;

    const int iBeg = ihalf * (IN_NODES / 2) + wave * (IN_NODES / 16);
    const int iN   = IN_NODES / 16;                    // 72

    v8f acc0 = {};
    v8f acc1 = {};
    // A[m][k] = x[m0+m, i, k]  (lane l -> row m=l; VGPR pair -> K = 2h, 2h+1)
    const float* xp0 = x + (size_t)(m0 + l) * (IN_NODES * IN_DIM) + (size_t)iBeg * IN_DIM + 2 * h;
    const float* xp1 = xp0 + (size_t)16 * (IN_NODES * IN_DIM);
    // B[k][d] = c[i,j] * W[i,j,d,k]  (lane l -> col d=l); shared by both tiles
    const float* wp = W + ((size_t)(iBeg * OUT_NODES + j) * OUT_DIM + l) * IN_DIM + 2 * h;
    const float* cp = c + iBeg * OUT_NODES + j;

    for (int it = 0; it < iN; ++it) {
        float cij = *cp;
        v2f b0  = { wp[0] * cij, wp[1] * cij };        // K = 0..3 slice
        v2f b1  = { wp[4] * cij, wp[5] * cij };        // K = 4..7 slice
        v2f a00 = { xp0[0], xp0[1] };
        v2f a01 = { xp0[4], xp0[5] };
        v2f a10 = { xp1[0], xp1[1] };
        v2f a11 = { xp1[4], xp1[5] };
        acc0 = __builtin_amdgcn_wmma_f32_16x16x4_f32(false, a00, false, b0, (short)0, acc0, false, false);
        acc0 = __builtin_amdgcn_wmma_f32_16x16x4_f32(false, a01, false, b1, (short)0, acc0, false, false);
        acc1 = __builtin_amdgcn_wmma_f32_16x16x4_f32(false, a10, false, b0, (short)0, acc1, false, false);
        acc1 = __builtin_amdgcn_wmma_f32_16x16x4_f32(false, a11, false, b1, (short)0, acc1, false, false);
        xp0 += IN_DIM;
        xp1 += IN_DIM;
        wp  += OUT_NODES * OUT_DIM * IN_DIM;
        cp  += OUT_NODES;
    }

    // D VGPR r = element [M = r + 8h][N = l]
    #pragma unroll
    for (int r = 0; r < 8; ++r) {
        red[wave * 512 +       (r + 8 * h) * 16 + l] = acc0[r];
        red[wave * 512 + 256 + (r + 8 * h) * 16 + l] = acc1[r];
    }
    __syncthreads();

    const int t = threadIdx.x;
    float v0 = 0.0f, v1 = 0.0f;
    #pragma unroll
    for (int w = 0; w < 8; ++w) { v0 += red[w * 512 + t]; v1 += red[w * 512 + 256 + t]; }
    const int row = t >> 4, col = t & 15;
    float* sp = s_part + (size_t)ihalf * SVOL + (size_t)j * (BATCH * OUT_DIM);
    sp[(m0 + row)      * OUT_DIM + col] = v0;
    sp[(m0 + 16 + row) * OUT_DIM + col] = v1;
}

// ---------------- squash (sums the two s partials; optional output write) ----------------
__global__ void k_squash(const float* __restrict__ s_part, float* __restrict__ v,
                         float* __restrict__ out, int write_out) {
    int r = blockIdx.x * blockDim.x + threadIdx.x;     // (j,b) row, 2560 total
    if (r >= OUT_NODES * BATCH) return;
    int j = r / BATCH, bb = r % BATCH;
    const float* sp0 = s_part +        (size_t)j * (BATCH * OUT_DIM) + bb * OUT_DIM;
    const float* sp1 = sp0 + SVOL;
    float vals[OUT_DIM];
    float sq = 0.0f;
    #pragma unroll
    for (int d = 0; d < OUT_DIM; ++d) { vals[d] = sp0[d] + sp1[d]; sq += vals[d] * vals[d]; }
    float scale = sq / (1.0f + sq) / sqrtf(sq);        // reference: sq/(1+sq) * s/||s||
    float* vp = v + (size_t)j * (BATCH * OUT_DIM) + bb * OUT_DIM;
    #pragma unroll
    for (int d = 0; d < OUT_DIM; ++d) {
        float val = vals[d] * scale;
        vp[d] = val;
        if (write_out) out[((size_t)bb * OUT_NODES + j) * OUT_DIM + d] = val;  // [b, j, d, 1]
    }
}

// ---------------- b[i,j] += (1/B) * sum_b u_hat[i,j,b,:] . v[j,b,:] ----------------
// G[(i_local,k), d] = sum_b x[b,i,k] * v[j,b,d] via WMMA; M packs 2 in-nodes x 8 k,
// K = 256 over batch.  Each wave owns TWO i-pair tiles so the v[j] B-operand is
// fetched once per 2 wmma.  Per-lane dot with W[i,j,d=l,:], shuffle-reduce over d.
// One wave per 4 in-nodes -> 2880 waves = 360 blocks.  No atomics.
__global__ __launch_bounds__(256) void k_bupd(const float* __restrict__ x,
                                              const float* __restrict__ W,
                                              const float* __restrict__ v,
                                              float* __restrict__ b) {
    const int wave = threadIdx.x >> 5;
    const int lane = threadIdx.x & 31;
    const int l    = lane & 15;
    const int h    = lane >> 4;
    const int t    = blockIdx.x * 8 + wave;            // 0..2879
    const int j    = t / (IN_NODES / 4);
    const int it   = t % (IN_NODES / 4);
    const int i0   = it * 4;                           // this wave covers i0..i0+3

    v8f acc0 = {};
    v8f acc1 = {};
    // A[M=l][K=kb+2h(+1)] = x[kb+2h, i, k];  addr = (kb+2h)*9216 + i0*8 + l  (lane-contiguous)
    const float* xp0 = x + (size_t)(2 * h) * (IN_NODES * IN_DIM) + i0 * IN_DIM + l;
    const float* xp1 = xp0 + 2 * IN_DIM;               // i0+2, i0+3 tile
    // B[K][N=l] = v[j, kb+K, l]; shared by both tiles
    const float* vp = v + (size_t)j * (BATCH * OUT_DIM) + (2 * h) * OUT_DIM + l;
    for (int kb = 0; kb < BATCH; kb += 4) {
        v2f bv = { vp[0],  vp[OUT_DIM] };
        v2f a0 = { xp0[0], xp0[IN_NODES * IN_DIM] };
        v2f a1 = { xp1[0], xp1[IN_NODES * IN_DIM] };
        acc0 = __builtin_amdgcn_wmma_f32_16x16x4_f32(false, a0, false, bv, (short)0, acc0, false, false);
        acc1 = __builtin_amdgcn_wmma_f32_16x16x4_f32(false, a1, false, bv, (short)0, acc1, false, false);
        xp0 += 4 * (IN_NODES * IN_DIM);
        xp1 += 4 * (IN_NODES * IN_DIM);
        vp  += 4 * OUT_DIM;
    }

    // lane (l,h), VGPR r of accT holds G[i = i0 + 2T + h, k = r, d = l]
    #pragma unroll
    for (int tile = 0; tile < 2; ++tile) {
        const int i = i0 + 2 * tile + h;
        const float* wr = W + ((size_t)(i * OUT_NODES + j) * OUT_DIM + l) * IN_DIM;
        v8f acc = tile ? acc1 : acc0;
        float dot = 0.0f;
        #pragma unroll
        for (int r = 0; r < 8; ++r) dot += acc[r] * wr[r];
        // reduce over d = l in 0..15 (xor masks <= 8 stay within each 16-lane half)
        dot += __shfl_xor(dot, 1, 32);
        dot += __shfl_xor(dot, 2, 32);
        dot += __shfl_xor(dot, 4, 32);
        dot += __shfl_xor(dot, 8, 32);
        if (l == 0) b[i * OUT_NODES + j] += dot * (1.0f / BATCH);
    }
}

extern "C" void kernel_launch(void* const* d_in, const int* in_sizes, int n_in,
                              void* d_out, int out_size, void* d_ws, size_t ws_size,
                              hipStream_t stream) {
    (void)in_sizes; (void)n_in; (void)out_size; (void)ws_size;
    const float* x = (const float*)d_in[0];   // [256, 1152, 8]
    const float* W = (const float*)d_in[1];   // [1152, 10, 16, 8]
    float* out = (float*)d_out;               // [256, 10, 16, 1]

    float* ws = (float*)d_ws;                 // ~583 KB total
    float* b  = ws;                                       // [1152,10]
    float* c  = b + IN_NODES * OUT_NODES;                 // [1152,10]
    float* sp = c + IN_NODES * OUT_NODES;                 // [2][10,256,16] partials
    float* v  = sp + 2 * SVOL;                            // [10,256,16]

    k_init<<<(IN_NODES * OUT_NODES + 255) / 256, 256, 0, stream>>>(b);
    for (int iter = 0; iter < 3; ++iter) {
        k_softmax<<<(IN_NODES + 255) / 256, 256, 0, stream>>>(b, c);
        k_s<<<160, 256, 0, stream>>>(x, W, c, sp);
        k_squash<<<(OUT_NODES * BATCH + 255) / 256, 256, 0, stream>>>(sp, v, out, iter == 2 ? 1 : 0);
        if (iter < 2)
            k_bupd<<<360, 256, 0, stream>>>(x, W, v, b);
    }
}